// MPDOPeriodic_59880434041480
// MI455X (gfx1250) — compile-verified
//
#include <hip/hip_runtime.h>
#include <math.h>

// Problem constants (from reference)
#define LSITES 64
#define DLOC   2
#define DBOND  8
#define CHIK   4
#define D2     64          // DBOND*DBOND
#define NTHREADS 128       // 4 waves of 32

typedef __attribute__((ext_vector_type(2))) float v2f;
typedef __attribute__((ext_vector_type(8))) float v8f;

// One workgroup per batch element b; edge (64x64 f32) chained through 64 sites.
//
// M is stored in a "paired-row" LDS layout so a WMMA B-operand is one
// contiguous ds_load_b64 per lane:
//   addr(row, col) = ((row>>1)*64 + col)*2 + (row&1)
// M is double-buffered: while site s is consumed, site s+1 is built
// (4 WMMAs from global b64 loads) into the other buffer -> 1 barrier/site.
__global__ __launch_bounds__(NTHREADS)
void mpdo_chain_kernel(const int* __restrict__ qn,
                       const float* __restrict__ tensors,
                       float* __restrict__ out)
{
    __shared__ float lds_M[2][D2 * D2];   // 2 x 16 KB transfer-matrix buffers
    __shared__ float lds_E[D2 * D2];      // 16 KB edge matrix (row-major)

    const int b    = blockIdx.x;
    const int tid  = threadIdx.x;
    const int wave = tid >> 5;            // owns edge rows 16*wave .. +15
    const int lane = tid & 31;
    const int half = lane >> 4;           // WMMA lane-half
    const int l16  = lane & 15;

    // edge <- Identity
    for (int e = tid; e < D2 * D2; e += NTHREADS)
        lds_E[e] = ((e >> 6) == (e & 63)) ? 1.0f : 0.0f;

    const int* qb = qn + (size_t)b * (2 * LSITES);
    const int rb = wave * 16;

    // Build transfer matrix of site ss into paired-row buffer Mb.
    //   P[(i*8+j),k] = Tr[(i*8+j)*4+k],  Q[k,(lp*8+mp)] = Tc[(lp*8+mp)*4+k]
    //   R = P@Q (WMMA 16x16x4) ; M[(i*8+lp)][(j*8+mp)] = R[(i*8+j)][(lp*8+mp)]
    auto buildM = [&](int ss, float* Mb) {
        const int ir = qb[ss];
        const int ic = qb[LSITES + ss];
        const float* Tr = tensors + (size_t)(ss * DLOC + ir) * (DBOND * DBOND * CHIK);
        const float* Tc = tensors + (size_t)(ss * DLOC + ic) * (DBOND * DBOND * CHIK);

        // A tile: lane l16 -> P row rb+l16, K = 2*half+{0,1} (k contiguous -> b64)
        v2f a_p = *(const v2f*)(Tr + (rb + l16) * CHIK + 2 * half);
        const int i_comp = wave * 2 + half;        // (R row)>>3, lane-constant
        #pragma unroll
        for (int t = 0; t < 4; ++t) {
            v2f b_q = *(const v2f*)(Tc + (t * 16 + l16) * CHIK + 2 * half);
            v8f c = {};
            c = __builtin_amdgcn_wmma_f32_16x16x4_f32(false, a_p, false, b_q,
                                                      (short)0, c, false, false);
            // Scatter: R row = wave*16 + 8*half + v => i = i_comp, j = v.
            const int colR = t * 16 + l16;
            const int lp = colR >> 3, mp = colR & 7;
            const int rowM = i_comp * 8 + lp;      // lane-constant per tile
            const int base = ((rowM >> 1) * 64 + mp) * 2 + (rowM & 1);
            #pragma unroll
            for (int v = 0; v < 8; ++v)
                Mb[base + v * 16] = c[v];          // col_M = v*8+mp -> stride 16
        }
    };

    buildM(0, lds_M[0]);
    __syncthreads();

    for (int s = 0; s < LSITES; ++s) {
        float* Mcur = lds_M[s & 1];
        float* Mnxt = lds_M[(s + 1) & 1];
        if (s + 1 < LSITES) buildM(s + 1, Mnxt);   // overlaps compute below

        // A tiles from lds_E (wave-private rows): b64, K = 4t + 2*half + {0,1}
        v2f A[16];
        #pragma unroll
        for (int t = 0; t < 16; ++t)
            A[t] = *(const v2f*)(&lds_E[(rb + l16) * 64 + 4 * t + 2 * half]);

        // 4 C tiles x 16 K-steps; B operand = contiguous b64 in paired layout
        v8f C[4];
        #pragma unroll
        for (int Jt = 0; Jt < 4; ++Jt) {
            v8f c = {};
            #pragma unroll
            for (int t = 0; t < 16; ++t) {
                v2f bm = *(const v2f*)(&Mcur[((2 * t + half) * 64 + Jt * 16 + l16) * 2]);
                c = __builtin_amdgcn_wmma_f32_16x16x4_f32(false, A[t], false, bm,
                                                          (short)0, c, false, false);
            }
            C[Jt] = c;
        }

        // Write updated edge rows back (wave-private, in-wave DS ordering)
        #pragma unroll
        for (int Jt = 0; Jt < 4; ++Jt) {
            #pragma unroll
            for (int v = 0; v < 8; ++v)
                lds_E[(rb + v + 8 * half) * 64 + Jt * 16 + l16] = C[Jt][v];
        }
        __syncthreads();   // seals: M(s+1) built, M(s) & edge reads done
    }

    // rho = trace(edge); out = log(complex(rho)) -> {log|rho|, arg}
    if (tid == 0) {
        float rho = 0.0f;
        for (int r = 0; r < D2; ++r) rho += lds_E[r * 64 + r];
        out[2 * b + 0] = __logf(fabsf(rho));
        out[2 * b + 1] = (rho < 0.0f) ? 3.14159265358979323846f : 0.0f;
    }
}

extern "C" void kernel_launch(void* const* d_in, const int* in_sizes, int n_in,
                              void* d_out, int out_size, void* d_ws, size_t ws_size,
                              hipStream_t stream) {
    const int*   qn      = (const int*)d_in[0];     // [B, 2L] int32
    const float* tensors = (const float*)d_in[1];   // [L, d, D, D, CHI] f32
    float*       out     = (float*)d_out;           // complex64 as {re,im} pairs

    const int B = in_sizes[0] / (2 * LSITES);       // 1024
    hipLaunchKernelGGL(mpdo_chain_kernel, dim3(B), dim3(NTHREADS), 0, stream,
                       qn, tensors, out);
}